// Local2Global_52810917871678
// MI455X (gfx1250) — compile-verified
//
#include <hip/hip_runtime.h>
#include <cstdint>

typedef __attribute__((ext_vector_type(2))) float v2f;
typedef __attribute__((ext_vector_type(8))) float v8f;
typedef __attribute__((__vector_size__(16))) int vld4i;  // async builtin pointee type

#define WMMA_F32(a, b, c) \
  __builtin_amdgcn_wmma_f32_16x16x4_f32(false, (a), false, (b), (short)0, (c), false, false)

constexpr int BS = 32, C_DIM = 512, HW = 4096;
constexpr int T_DIM = 64, TOK = 128, NH = 2, CH = 256;
constexpr int NS = 64, SCHUNK = 512, NCHUNK = HW / SCHUNK;   // 8 chunks
constexpr int NT = SCHUNK / NS;                              // 8 s-tiles per WG
constexpr int KSTR = 68, ESTR = 68;                          // LDS row pads
constexpr int KH = 128 * KSTR;                               // half K tile (rows 0-127)
constexpr float SCALE = 0.0625f;                             // ch^-0.5
constexpr float EPS = 1e-5f;

// ---------------- async global->LDS (gfx1250) with sync fallback -----------
#if defined(__has_builtin)
#if __has_builtin(__builtin_amdgcn_global_load_async_to_lds_b128)
#define HAS_ASYNC_LDS 1
#endif
#if __has_builtin(__builtin_amdgcn_s_wait_asynccnt)
#define HAS_WAIT_ASYNC 1
#endif
#endif

#ifndef HAS_ASYNC_LDS
#warning "gfx1250 async-to-LDS builtin NOT available; using synchronous copy fallback"
#endif

__device__ __forceinline__ void async_cp16(const float* g, float* l) {
#ifdef HAS_ASYNC_LDS
  __builtin_amdgcn_global_load_async_to_lds_b128(
      (vld4i*)(uintptr_t)g, (vld4i*)l, 0, 0);
#else
  *(float4*)l = *(const float4*)g;
#endif
}

template <int N>
__device__ __forceinline__ void wait_asynccnt() {
#ifdef HAS_ASYNC_LDS
#ifdef HAS_WAIT_ASYNC
  __builtin_amdgcn_s_wait_asynccnt(N);
#else
  if constexpr (N == 0) asm volatile("s_wait_asynccnt 0x0" ::: "memory");
  else                  asm volatile("s_wait_asynccnt 0x10" ::: "memory");
#endif
#endif
}

// ---------------------------------------------------------------------------
// Kernel 1: q projection.  qT[b][h][c][t] = q_b[co] + tokens[t][b][:]·q_w[co][:]
// ---------------------------------------------------------------------------
__global__ __launch_bounds__(256) void qproj_kernel(
    const float* __restrict__ tokens, const float* __restrict__ q_w,
    const float* __restrict__ q_b, float* __restrict__ qT) {
  __shared__ float tok[16 * 128];
  __shared__ float qw[128 * 129];
  const int b = blockIdx.x;
  const int t0 = blockIdx.y * 16;
  const int tid = threadIdx.x;

  for (int k = 0; k < 8; ++k) {
    int i = tid + k * 256;
    int tl = i >> 7, d = i & 127;
    tok[i] = tokens[((size_t)(t0 + tl) * BS + b) * TOK + d];
  }
  const int co_l = tid & 127;
  const int th = tid >> 7;
  for (int chunk = 0; chunk < 4; ++chunk) {
    __syncthreads();
    for (int k = 0; k < 64; ++k) {
      int i = tid + k * 256;
      int r = i >> 7, d = i & 127;
      qw[r * 129 + d] = q_w[(size_t)(chunk * 128 + r) * TOK + d];
    }
    __syncthreads();
    int co = chunk * 128 + co_l;
    float bias = q_b[co];
    float acc[8];
#pragma unroll
    for (int ti = 0; ti < 8; ++ti) acc[ti] = bias;
    for (int d = 0; d < 128; ++d) {
      float wv = qw[co_l * 129 + d];
#pragma unroll
      for (int ti = 0; ti < 8; ++ti) acc[ti] += tok[(th * 8 + ti) * 128 + d] * wv;
    }
    int h = co >> 8, c = co & 255;
    float* outp = qT + ((size_t)(b * NH + h) * CH + c) * T_DIM + t0 + th * 8;
#pragma unroll
    for (int ti = 0; ti < 8; ++ti) outp[ti] = acc[ti];
  }
}

// ---------------------------------------------------------------------------
// Kernel 2: fused attention per (b,h,s-chunk). Double-buffered async K tiles.
// K tile stored as two 128-row halves so all DS offsets fit 16-bit immediates.
// ---------------------------------------------------------------------------
__device__ __forceinline__ void issue_tile(const float* fp, float* ka, float* kb,
                                           int tid) {
#pragma unroll
  for (int p = 0; p < 16; ++p) {
    int i = tid + p * 256;       // float4 index 0..4095
    int c = i >> 4, j = i & 15;  // row 0..255 (p<8 -> row<128), 16B col 0..15
    float* dst = (p < 8) ? (ka + c * KSTR + j * 4) : (kb + (c - 128) * KSTR + j * 4);
    async_cp16(fp + (size_t)c * HW + j * 4, dst);
  }
}

__global__ __launch_bounds__(256) void attn_kernel(
    const float* __restrict__ features, const float* __restrict__ qT,
    float* __restrict__ attn_out, float* __restrict__ Obuf) {
  __shared__ float K0a[KH], K0b[KH];  // buffer 0: rows 0-127 / 128-255
  __shared__ float K1a[KH], K1b[KH];  // buffer 1
  __shared__ float Elds[T_DIM * ESTR];

  const int chunk = blockIdx.x;
  const int bh = blockIdx.y;
  const int tid = threadIdx.x;
  const int lane = tid & 31, wave = tid >> 5;
  const int wm = wave >> 1, wn = wave & 1;
  const int lo = lane & 15, hi = lane >> 4;

  const float* fbase = features + (size_t)bh * CH * HW + chunk * SCHUNK;
  float* attnp = attn_out + (size_t)bh * T_DIM * HW + chunk * SCHUNK;

  // kick off first K tile before doing anything else
  issue_tile(fbase, K0a, K0b, tid);

  // Q A-fragments (loop-invariant): 16 rows x 256 ch per wave, 128 VGPRs
  const float* qg = qT + (size_t)bh * CH * T_DIM + 16 * wm + lo;
  v2f qa[64];
#pragma unroll
  for (int cs = 0; cs < 64; ++cs) {
    qa[cs].x = qg[(4 * cs + 2 * hi) * T_DIM];
    qa[cs].y = qg[(4 * cs + 2 * hi + 1) * T_DIM];
  }

  v8f oacc[8];
#pragma unroll
  for (int nn = 0; nn < 8; ++nn)
    oacc[nn] = (v8f){0.f, 0.f, 0.f, 0.f, 0.f, 0.f, 0.f, 0.f};

  for (int it = 0; it < NT; ++it) {
    float* Ka = (it & 1) ? K1a : K0a;
    float* Kb = (it & 1) ? K1b : K0b;
    if (it + 1 < NT) {
      issue_tile(fbase + (it + 1) * NS, (it & 1) ? K0a : K1a, (it & 1) ? K0b : K1b,
                 tid);
      wait_asynccnt<16>();  // loads retire in order: previous 16 (= tile it) done
    } else {
      wait_asynccnt<0>();
    }
    __syncthreads();

    // ---- GEMM1: S[16x32 per wave] = q[16x256] x K[256x32] ----
    v8f s0v = (v8f){0.f, 0.f, 0.f, 0.f, 0.f, 0.f, 0.f, 0.f};
    v8f s1v = (v8f){0.f, 0.f, 0.f, 0.f, 0.f, 0.f, 0.f, 0.f};
    const int col = 32 * wn + lo;
    // Launder the +16 column offset so the two B-operand address chains use
    // provably-unrelated registers: the load-store optimizer can then only
    // pair {+0,+KSTR} (the two halves of ONE wmma operand) into a single
    // ds_load_2addr with an adjacent destination pair -> no repacking movs.
    int col16 = col + 16;
    asm("" : "+v"(col16));
    const float* Ka0 = Ka + 2 * hi * KSTR + col;
    const float* Ka1 = Ka + 2 * hi * KSTR + col16;
    const float* Kb0 = Kb + 2 * hi * KSTR + col;
    const float* Kb1 = Kb + 2 * hi * KSTR + col16;
#pragma unroll
    for (int cs = 0; cs < 64; ++cs) {
      const float* r0 = (cs < 32) ? (Ka0 + 4 * cs * KSTR) : (Kb0 + 4 * (cs - 32) * KSTR);
      const float* r1 = (cs < 32) ? (Ka1 + 4 * cs * KSTR) : (Kb1 + 4 * (cs - 32) * KSTR);
      v2f b0, b1;
      b0.x = r0[0];
      b0.y = r0[KSTR];
      b1.x = r1[0];
      b1.y = r1[KSTR];
      s0v = WMMA_F32(qa[cs], b0, s0v);
      s1v = WMMA_F32(qa[cs], b1, s1v);
    }
    // scale, emit attn output, exp -> Elds
    const int s0 = it * NS;
#pragma unroll
    for (int r = 0; r < 8; ++r) {
      int row = 16 * wm + 8 * hi + r;
      float sA = s0v[r] * SCALE, sB = s1v[r] * SCALE;
      attnp[(size_t)row * HW + s0 + col] = sA;
      attnp[(size_t)row * HW + s0 + col + 16] = sB;
      Elds[row * ESTR + col] = __expf(sA);
      Elds[row * ESTR + col + 16] = __expf(sB);
    }
    __syncthreads();

    // ---- GEMM2: O[16x128 per wave] += E[16x64] x K^T[64x128] ----
    const int m = 16 * wm + lo;
    const float* K2Base = (wn ? Kb : Ka) + lo * KSTR;  // wave's 128-ch half
#pragma unroll
    for (int ks = 0; ks < 16; ++ks) {
      int k = 4 * ks + 2 * hi;
      v2f av = *(const v2f*)&Elds[m * ESTR + k];
#pragma unroll
      for (int nn = 0; nn < 8; ++nn) {
        v2f bv = *(const v2f*)&K2Base[16 * nn * KSTR + k];
        oacc[nn] = WMMA_F32(av, bv, oacc[nn]);
      }
    }
    __syncthreads();  // all waves done with K/E before overwrite
  }

  // accumulate unnormalized O partial (chunks combine by pure addition)
  float* op = Obuf + (size_t)bh * T_DIM * CH;
#pragma unroll
  for (int nn = 0; nn < 8; ++nn) {
#pragma unroll
    for (int r = 0; r < 8; ++r) {
      int row = 16 * wm + 8 * hi + r;
      int ch = 128 * wn + 16 * nn + lo;
      unsafeAtomicAdd(&op[row * CH + ch], oacc[nn][r]);
    }
  }
}

// ---------------------------------------------------------------------------
// Kernel 3: l = sum exp(attn), normalize O, project 512->128, residual, LN.
// ---------------------------------------------------------------------------
__global__ __launch_bounds__(256) void out_kernel(
    const float* __restrict__ tokens, const float* __restrict__ proj_w,
    const float* __restrict__ proj_b, const float* __restrict__ ln_w,
    const float* __restrict__ ln_b, const float* __restrict__ attn,
    const float* __restrict__ Obuf, float* __restrict__ out) {
  __shared__ float l_lds[16];
  __shared__ float v_lds[8 * 512];
  __shared__ float wtile[128 * 129];
  __shared__ float sum1[8], sum2[8];

  const int b = blockIdx.x;
  const int t0 = blockIdx.y * 8;
  const int tid = threadIdx.x;

  if (tid < 16) l_lds[tid] = 0.f;
  __syncthreads();
  {  // softmax denominators (max-free; logits ~N(0,1), no overflow risk in fp32)
    int row = tid >> 4, l16 = tid & 15;
    int ti = row >> 1, h = row & 1;
    const float* ap = attn + ((size_t)(b * NH + h) * T_DIM + t0 + ti) * HW;
    float s = 0.f;
    for (int k = 0; k < HW / 16; ++k) s += __expf(ap[l16 + 16 * k]);
    unsafeAtomicAdd(&l_lds[row], s);
  }
  __syncthreads();
  for (int p = 0; p < 16; ++p) {
    int flat = tid + p * 256;
    int ti = flat >> 9, co = flat & 511;
    int h = co >> 8, c = co & 255;
    v_lds[flat] =
        Obuf[((size_t)(b * NH + h) * T_DIM + t0 + ti) * CH + c] / l_lds[ti * 2 + h];
  }

  const int d = tid & 127, dh = tid >> 7;
  float acc[4] = {0.f, 0.f, 0.f, 0.f};
  for (int chunk = 0; chunk < 4; ++chunk) {
    __syncthreads();
    for (int k = 0; k < 64; ++k) {
      int i = tid + k * 256;
      int r = i >> 7, cc = i & 127;
      wtile[r * 129 + cc] = proj_w[(size_t)r * C_DIM + chunk * 128 + cc];
    }
    __syncthreads();
    for (int cc = 0; cc < 128; ++cc) {
      float wv = wtile[d * 129 + cc];
      int co = chunk * 128 + cc;
#pragma unroll
      for (int ti = 0; ti < 4; ++ti) acc[ti] += v_lds[(dh * 4 + ti) * 512 + co] * wv;
    }
  }
  __syncthreads();
  if (tid < 8) { sum1[tid] = 0.f; sum2[tid] = 0.f; }
  __syncthreads();

  float tokv[4];
  float pb = proj_b[d];
#pragma unroll
  for (int ti = 0; ti < 4; ++ti) {
    int t = t0 + dh * 4 + ti;
    tokv[ti] = tokens[((size_t)t * BS + b) * TOK + d] + acc[ti] + pb;
    float s = tokv[ti], s2 = tokv[ti] * tokv[ti];
    for (int off = 16; off > 0; off >>= 1) {
      s += __shfl_down(s, off);
      s2 += __shfl_down(s2, off);
    }
    if ((tid & 31) == 0) {
      unsafeAtomicAdd(&sum1[dh * 4 + ti], s);
      unsafeAtomicAdd(&sum2[dh * 4 + ti], s2);
    }
  }
  __syncthreads();
  float lw = ln_w[d], lb = ln_b[d];
#pragma unroll
  for (int ti = 0; ti < 4; ++ti) {
    int t = t0 + dh * 4 + ti;
    float mu = sum1[dh * 4 + ti] * (1.f / 128.f);
    float var = sum2[dh * 4 + ti] * (1.f / 128.f) - mu * mu;
    out[((size_t)t * BS + b) * TOK + d] =
        (tokv[ti] - mu) * rsqrtf(var + EPS) * lw + lb;
  }
}

// ---------------------------------------------------------------------------
extern "C" void kernel_launch(void* const* d_in, const int* in_sizes, int n_in,
                              void* d_out, int out_size, void* d_ws, size_t ws_size,
                              hipStream_t stream) {
  (void)in_sizes; (void)n_in; (void)out_size; (void)ws_size;
  const float* features = (const float*)d_in[0];
  const float* tokens   = (const float*)d_in[1];
  const float* q_w      = (const float*)d_in[2];
  const float* q_b      = (const float*)d_in[3];
  const float* proj_w   = (const float*)d_in[4];
  const float* proj_b   = (const float*)d_in[5];
  const float* ln_w     = (const float*)d_in[6];
  const float* ln_b     = (const float*)d_in[7];

  float* out_tok  = (float*)d_out;
  float* out_attn = out_tok + (size_t)T_DIM * BS * TOK;

  float* ws_q = (float*)d_ws;                         // [bs][nh][ch][T], 4MB
  float* ws_O = ws_q + (size_t)BS * NH * CH * T_DIM;  // [bs][nh][T][ch], 4MB

  (void)hipMemsetAsync(ws_O, 0, (size_t)BS * NH * T_DIM * CH * sizeof(float), stream);
  qproj_kernel<<<dim3(BS, 4), 256, 0, stream>>>(tokens, q_w, q_b, ws_q);
  attn_kernel<<<dim3(NCHUNK, BS * NH), 256, 0, stream>>>(features, ws_q, out_attn, ws_O);
  out_kernel<<<dim3(BS, 8), 256, 0, stream>>>(tokens, proj_w, proj_b, ln_w, ln_b,
                                              out_attn, ws_O, out_tok);
}